// my_conv_22308060136095
// MI455X (gfx1250) — compile-verified
//
#include <hip/hip_runtime.h>

// Depthwise 3x3 conv, 2 filters/channel, interleaved output channels.
// B=32, C=192, H=W=56, pad=1, stride=1.  ~3 FLOP/byte -> pure HBM-bandwidth
// bound (floor ~10us at 23.3 TB/s).  Built around the CDNA5 async
// global->LDS data mover: each input plane streams to LDS exactly once
// (flat copy, INST_OFFSET applies to both LDS and memory address), the 3x3
// halo reuse (18 FMAs/loaded float) comes from LDS, and both HBM streams
// use non-temporal hints (231 MB footprint > 192 MB L2).

typedef float v4f __attribute__((ext_vector_type(4)));

#define B_  32
#define C_  192
#define H_  56
#define W_  56
#define HW_ (H_ * W_)        // 3136 pixels = 12,544 B per plane
#define NG_ (HW_ / 4)        // 784 aligned float4 groups per plane

__global__ __launch_bounds__(256)
void dwconv2_async_kernel(const float* __restrict__ x,
                          const float* __restrict__ w1g,
                          const float* __restrict__ w2g,
                          const float* __restrict__ bias,
                          float* __restrict__ out)
{
    // Unpadded: with row stride == W, 32 consecutive pixels are 32 contiguous
    // LDS floats even across a row boundary -> conflict-free on 64 banks.
    __shared__ __align__(16) float smem[HW_];

    const int blk = blockIdx.x;          // one block per (b, c) plane
    const int b   = blk / C_;
    const int c   = blk - b * C_;
    const int tid = threadIdx.x;

    const float* xplane = x + (size_t)(b * C_ + c) * HW_;

    // ---- flat async copy: 784 x 16 B, 4 instructions, no address math ----
    // ISA: the instruction offset is added to both the LDS and global address,
    // so one (lds, global) base pair + offset: immediates covers the plane.
    unsigned l = (unsigned)(uintptr_t)(void*)smem + (unsigned)tid * 16u;
    unsigned long long g = (unsigned long long)(uintptr_t)xplane
                         + (unsigned long long)tid * 16ull;
    asm volatile("global_load_async_to_lds_b128 %0, %1, off th:TH_LOAD_NT"
                 :: "v"(l), "v"(g) : "memory");
    asm volatile("global_load_async_to_lds_b128 %0, %1, off offset:4096 th:TH_LOAD_NT"
                 :: "v"(l), "v"(g) : "memory");
    asm volatile("global_load_async_to_lds_b128 %0, %1, off offset:8192 th:TH_LOAD_NT"
                 :: "v"(l), "v"(g) : "memory");
    if (tid < 16)   // tail: 12,544 - 12,288 = 256 B
        asm volatile("global_load_async_to_lds_b128 %0, %1, off offset:12288 th:TH_LOAD_NT"
                     :: "v"(l), "v"(g) : "memory");

    // Workgroup-uniform weights/bias (c from blockIdx -> scalar loads),
    // issued before the async wait so SMEM overlaps the DMA.
    float w1[9], w2[9];
#pragma unroll
    for (int k = 0; k < 9; ++k) {
        w1[k] = w1g[c * 9 + k];
        w2[k] = w2g[c * 9 + k];
    }
    const float b1 = bias[2 * c + 0];
    const float b2 = bias[2 * c + 1];

    asm volatile("s_wait_asynccnt 0x0" ::: "memory");
    __syncthreads();

    float* o1 = out + (size_t)(b * 2 * C_ + 2 * c) * HW_;  // filter-1 plane
    float* o2 = o1 + HW_;                                  // filter-2 plane

    // Each thread computes an aligned group of 4 pixels (56 % 4 == 0, so a
    // group never straddles a row): 9 DS loads -> 72 FMAs -> two 16 B NT
    // stores.  One divide total; row/col updated incrementally (256 = 18*14+4).
    int r   = tid / 14;          // 14 groups per row
    int c14 = tid - r * 14;

    const v4f zero4 = {0.f, 0.f, 0.f, 0.f};

#pragma unroll
    for (int it = 0; it < 4; ++it) {
        int grp = tid + it * 256;
        if (grp < NG_) {
            int col0 = c14 * 4;
            float a1[4] = {b1, b1, b1, b1};
            float a2[4] = {b2, b2, b2, b2};
#pragma unroll
            for (int kh = 0; kh < 3; ++kh) {
                int  rr  = r + kh - 1;
                bool rin = (unsigned)rr < (unsigned)H_;
                int  base = rin ? rr * W_ + col0 : 0;     // safe clamped addr
                v4f  m = *(const v4f*)(smem + base);      // ds_load_b128
                m = rin ? m : zero4;
                bool lin = rin && (col0 > 0);
                bool rgt = rin && (col0 < W_ - 4);
                float L = smem[lin ? base - 1 : 0];       // ds_load_b32
                float R = smem[rgt ? base + 4 : 0];       // ds_load_b32
                L = lin ? L : 0.0f;
                R = rgt ? R : 0.0f;
                float win[6] = {L, m.x, m.y, m.z, m.w, R};
#pragma unroll
                for (int j = 0; j < 4; ++j)
#pragma unroll
                    for (int kw = 0; kw < 3; ++kw) {
                        float v = win[j + kw];
                        a1[j] = fmaf(v, w1[kh * 3 + kw], a1[j]);
                        a2[j] = fmaf(v, w2[kh * 3 + kw], a2[j]);
                    }
            }
            v4f s1 = {a1[0], a1[1], a1[2], a1[3]};
            v4f s2 = {a2[0], a2[1], a2[2], a2[3]};
            // Write-once output stream: bypass near caches.
            __builtin_nontemporal_store(s1, (v4f*)(o1 + (size_t)grp * 4));
            __builtin_nontemporal_store(s2, (v4f*)(o2 + (size_t)grp * 4));
        }
        r += 18; c14 += 4;
        if (c14 >= 14) { c14 -= 14; ++r; }
    }
}

extern "C" void kernel_launch(void* const* d_in, const int* in_sizes, int n_in,
                              void* d_out, int out_size, void* d_ws, size_t ws_size,
                              hipStream_t stream) {
    (void)in_sizes; (void)n_in; (void)d_ws; (void)ws_size; (void)out_size;
    const float* x  = (const float*)d_in[0];
    const float* w1 = (const float*)d_in[1];
    const float* w2 = (const float*)d_in[2];
    const float* bs = (const float*)d_in[3];
    float* out = (float*)d_out;

    dim3 grid(B_ * C_);   // 6144 workgroups, 256 threads = 8 wave32 each
    dwconv2_async_kernel<<<grid, 256, 0, stream>>>(x, w1, w2, bs, out);
}